// GCNBackbone_68676527063443
// MI455X (gfx1250) — compile-verified
//
#include <hip/hip_runtime.h>
#include <math.h>

typedef __attribute__((ext_vector_type(2))) float v2f;
typedef __attribute__((ext_vector_type(8))) float v8f;

#define FDIM 128  // IN_DIM == HID == 128

// ---------------------------------------------------------------------------
// Utility: zero a float buffer
// ---------------------------------------------------------------------------
__global__ __launch_bounds__(256) void zero_f32(float* __restrict__ p, size_t n) {
    size_t i = (size_t)blockIdx.x * blockDim.x + threadIdx.x;
    if (i < n) p[i] = 0.0f;
}

// ---------------------------------------------------------------------------
// Degree accumulation: deg[dst[e]] += 1  (float, native global_atomic_add_f32)
// ---------------------------------------------------------------------------
__global__ __launch_bounds__(256) void deg_accum(const int* __restrict__ dst,
                                                 float* __restrict__ deg, int nEdges) {
    int e = blockIdx.x * blockDim.x + threadIdx.x;
    if (e < nEdges) {
        __hip_atomic_fetch_add(&deg[dst[e]], 1.0f,
                               __ATOMIC_RELAXED, __HIP_MEMORY_SCOPE_AGENT);
    }
}

// dinv[i] = rsqrt(deg[i] + 1)   (+1 = self loop; degree always > 0)
__global__ __launch_bounds__(256) void deg_to_dinv(float* __restrict__ deg, int n) {
    int i = blockIdx.x * blockDim.x + threadIdx.x;
    if (i < n) deg[i] = rsqrtf(deg[i] + 1.0f);
}

// ---------------------------------------------------------------------------
// H[N,128] = X[N,128] @ W[128,128]   via V_WMMA_F32_16X16X4_F32
// One wave per 16-row tile; each wave computes the full 16x128 output slab
// (8 column tiles, 8 v8f accumulators). Guard is wave-uniform -> EXEC all-1s.
//
// fp32 WMMA layouts (wave32):
//   A 16x4:  lanes 0-15 hold M=lane, VGPR{0,1}=K{0,1}; lanes 16-31 K{2,3}
//   B 4x16:  lanes 0-15 hold N=lane, VGPR{0,1}=K{0,1}; lanes 16-31 K{2,3}
//   C/D 16x16: VGPR i -> M = i + 8*(lane>=16), N = lane&15
// ---------------------------------------------------------------------------
__global__ __launch_bounds__(256) void gemm_xw_wmma(const float* __restrict__ X,
                                                    const float* __restrict__ W,
                                                    float* __restrict__ H,
                                                    int nTiles) {
    int wave = (int)((blockIdx.x * (unsigned)blockDim.x + threadIdx.x) >> 5);
    if (wave >= nTiles) return;                 // uniform across the wave
    int lane = threadIdx.x & 31;
    int m  = lane & 15;
    int hi = lane >> 4;                         // 0: K pair {0,1}, 1: K pair {2,3}

    const float* xrow = X + ((size_t)wave * 16 + m) * FDIM;

    v8f acc[8] = {};                            // 8 column tiles of 16

    for (int k = 0; k < FDIM; k += 4) {
        int kA = k + 2 * hi;
        v2f a;
        a.x = xrow[kA];
        a.y = xrow[kA + 1];
        const float* wp = W + (size_t)kA * FDIM + m;
#pragma unroll
        for (int c = 0; c < 8; ++c) {
            v2f b;
            b.x = wp[c * 16];
            b.y = wp[c * 16 + FDIM];
            acc[c] = __builtin_amdgcn_wmma_f32_16x16x4_f32(
                /*neg_a=*/false, a, /*neg_b=*/false, b,
                /*c_mod=*/(short)0, acc[c], /*reuse_a=*/false, /*reuse_b=*/false);
        }
    }

    // store D: row = wave*16 + hi*8 + i, col = c*16 + m
    float* hout = H + ((size_t)wave * 16 + hi * 8) * FDIM + m;
#pragma unroll
    for (int c = 0; c < 8; ++c) {
#pragma unroll
        for (int i = 0; i < 8; ++i) {
            hout[(size_t)i * FDIM + c * 16] = acc[c][i];
        }
    }
}

// ---------------------------------------------------------------------------
// Edge scatter: one wave per edge, float4 per lane (32*4 = 128 features)
// Out[dst] += H[src] * dinv[src]*dinv[dst]
// ---------------------------------------------------------------------------
__global__ __launch_bounds__(256) void scatter_edges(const float* __restrict__ H,
                                                     const float* __restrict__ dinv,
                                                     const int* __restrict__ src,
                                                     const int* __restrict__ dst,
                                                     float* __restrict__ Out,
                                                     int nEdges) {
    size_t t = (size_t)blockIdx.x * blockDim.x + threadIdx.x;
    int e = (int)(t >> 5);
    if (e >= nEdges) return;
    int lane = (int)(t & 31);

    int s = src[e];
    int d = dst[e];
    float nrm = dinv[s] * dinv[d];

    const float4* hv = (const float4*)(H + (size_t)s * FDIM) + lane;
    float4 v = *hv;

    float* o = Out + (size_t)d * FDIM + lane * 4;
    __hip_atomic_fetch_add(o + 0, v.x * nrm, __ATOMIC_RELAXED, __HIP_MEMORY_SCOPE_AGENT);
    __hip_atomic_fetch_add(o + 1, v.y * nrm, __ATOMIC_RELAXED, __HIP_MEMORY_SCOPE_AGENT);
    __hip_atomic_fetch_add(o + 2, v.z * nrm, __ATOMIC_RELAXED, __HIP_MEMORY_SCOPE_AGENT);
    __hip_atomic_fetch_add(o + 3, v.w * nrm, __ATOMIC_RELAXED, __HIP_MEMORY_SCOPE_AGENT);
}

// ---------------------------------------------------------------------------
// Finalize (in place): Acc[i] += H[i]*dinv[row]^2 (self loop) + bias[col];
// optional ReLU.
// ---------------------------------------------------------------------------
__global__ __launch_bounds__(256) void finalize_layer(float* __restrict__ Acc,
                                                      const float* __restrict__ Hself,
                                                      const float* __restrict__ dinv,
                                                      const float* __restrict__ bias,
                                                      size_t nElems, int doRelu) {
    size_t i = (size_t)blockIdx.x * blockDim.x + threadIdx.x;
    if (i >= nElems) return;
    size_t row = i >> 7;          // /128
    int col = (int)(i & (FDIM - 1));
    float dv = dinv[row];
    float v = Acc[i] + Hself[i] * dv * dv + bias[col];
    if (doRelu) v = fmaxf(v, 0.0f);
    Acc[i] = v;
}

// ---------------------------------------------------------------------------
// Host-side launch
// ---------------------------------------------------------------------------
extern "C" void kernel_launch(void* const* d_in, const int* in_sizes, int n_in,
                              void* d_out, int out_size, void* d_ws, size_t ws_size,
                              hipStream_t stream) {
    const float* x  = (const float*)d_in[0];
    const int*   ei = (const int*)d_in[1];
    const float* W1 = (const float*)d_in[2];
    const float* b1 = (const float*)d_in[3];
    const float* W2 = (const float*)d_in[4];
    const float* b2 = (const float*)d_in[5];
    float* out = (float*)d_out;

    const int N = in_sizes[0] / FDIM;     // 100000
    const int E = in_sizes[1] / 2;        // 1600000
    const int* src = ei;
    const int* dst = ei + E;

    // workspace layout: dinv[N] | h[N*128] | agg[N*128]
    float* dinv = (float*)d_ws;
    float* h    = dinv + (size_t)N;
    float* agg  = h + (size_t)N * FDIM;

    const size_t feat   = (size_t)N * FDIM;
    const int    nTiles = (N + 15) / 16;  // N is a multiple of 16 here

    dim3 blk(256);
    auto cdiv = [](size_t a, size_t b) { return (unsigned)((a + b - 1) / b); };

    // --- symmetric normalization factors ---
    zero_f32<<<cdiv((size_t)N, 256), blk, 0, stream>>>(dinv, (size_t)N);
    deg_accum<<<cdiv((size_t)E, 256), blk, 0, stream>>>(dst, dinv, E);
    deg_to_dinv<<<cdiv((size_t)N, 256), blk, 0, stream>>>(dinv, N);

    // --- layer 1: h = x @ W1 ; agg = A_norm * h ; agg = relu(agg + b1) ---
    gemm_xw_wmma<<<cdiv((size_t)nTiles * 32, 256), blk, 0, stream>>>(x, W1, h, nTiles);
    zero_f32<<<cdiv(feat, 256), blk, 0, stream>>>(agg, feat);
    scatter_edges<<<cdiv((size_t)E * 32, 256), blk, 0, stream>>>(h, dinv, src, dst, agg, E);
    finalize_layer<<<cdiv(feat, 256), blk, 0, stream>>>(agg, h, dinv, b1, feat, /*relu=*/1);

    // --- layer 2: h = agg @ W2 ; out = A_norm * h + b2 ---
    gemm_xw_wmma<<<cdiv((size_t)nTiles * 32, 256), blk, 0, stream>>>(agg, W2, h, nTiles);
    zero_f32<<<cdiv(feat, 256), blk, 0, stream>>>(out, feat);
    scatter_edges<<<cdiv((size_t)E * 32, 256), blk, 0, stream>>>(h, dinv, src, dst, out, E);
    finalize_layer<<<cdiv(feat, 256), blk, 0, stream>>>(out, h, dinv, b2, feat, /*relu=*/0);
}